// DeciLMSSMDecoderLayer_55817394979023
// MI455X (gfx1250) — compile-verified
//
#include <hip/hip_runtime.h>
#include <hip/hip_bf16.h>
#include <math.h>

typedef __attribute__((ext_vector_type(2))) float v2f;
typedef __attribute__((ext_vector_type(8))) float v8f;

#define L_SEQ    1024
#define DMODEL   2048
#define H_HEADS  32
#define P_DIM    64
#define G_GRP    8
#define S_STATE  32
#define D_SSM    2048            // H*P
#define CONVDIM  2560            // D_SSM + 2*G*S
#define PROJDIM  4640            // 2*D_SSM + 2*G*S + H
#define KCONV    4

// ---------------------------------------------------------------------------
// f32 WMMA GEMM:  C[M,N] = (A[M,K] @ B[K,N] + bias[N]) * scale[N]
// block = 128 threads (4 waves), block tile 64(M) x 128(N).
// Each wave computes 64(M) x 32(N): 4 M-tiles x 2 N-tiles = 8 accumulators.
// Per K-step of 4: 4 x b64 A loads + 4 x b32 B loads feed 8 WMMAs
// (1.0 wmma/load vs 0.4 before; A frags reused 2x, B frags reused 4x).
// REQUIRES N % 32 == 0 so the n0 guard stays wave-uniform (EXEC all-ones
// on the WMMA path, per ISA restriction). Both call sites satisfy this.
// ---------------------------------------------------------------------------
__global__ void gemm_wmma_f32(const float* __restrict__ A,
                              const float* __restrict__ B,
                              const float* __restrict__ bias,
                              const float* __restrict__ scale,
                              float* __restrict__ C,
                              int M, int N, int K)
{
    const int lane = threadIdx.x & 31;
    const int wave = threadIdx.x >> 5;
    const int half = lane >> 4;      // 0: K=0,1  1: K=2,3
    const int r    = lane & 15;      // M (for A) / N (for B,C,D) within tile

    const int n0 = blockIdx.x * 128 + wave * 32;
    const int m0 = blockIdx.y * 64;
    if (n0 >= N) return;             // wave-uniform (N % 32 == 0)

    v8f z = {0.f,0.f,0.f,0.f,0.f,0.f,0.f,0.f};
    v8f acc00 = z, acc01 = z, acc10 = z, acc11 = z;
    v8f acc20 = z, acc21 = z, acc30 = z, acc31 = z;

    const float* Arow0 = A + (size_t)(m0 +  0 + r) * K;
    const float* Arow1 = A + (size_t)(m0 + 16 + r) * K;
    const float* Arow2 = A + (size_t)(m0 + 32 + r) * K;
    const float* Arow3 = A + (size_t)(m0 + 48 + r) * K;
    const float* Bcol0 = B + n0 +  0 + r;
    const float* Bcol1 = B + n0 + 16 + r;

    #pragma unroll 2
    for (int k0 = 0; k0 < K; k0 += 4) {
        const int ka = k0 + 2 * half;
        v2f b0, b1;
        b0.x = Bcol0[(size_t)(ka    ) * N];
        b0.y = Bcol0[(size_t)(ka + 1) * N];
        b1.x = Bcol1[(size_t)(ka    ) * N];
        b1.y = Bcol1[(size_t)(ka + 1) * N];
        v2f a0 = *(const v2f*)(Arow0 + ka);   // 8B-aligned (ka even)
        v2f a1 = *(const v2f*)(Arow1 + ka);
        v2f a2 = *(const v2f*)(Arow2 + ka);
        v2f a3 = *(const v2f*)(Arow3 + ka);
        acc00 = __builtin_amdgcn_wmma_f32_16x16x4_f32(false, a0, false, b0, (short)0, acc00, false, false);
        acc10 = __builtin_amdgcn_wmma_f32_16x16x4_f32(false, a1, false, b0, (short)0, acc10, false, false);
        acc20 = __builtin_amdgcn_wmma_f32_16x16x4_f32(false, a2, false, b0, (short)0, acc20, false, false);
        acc30 = __builtin_amdgcn_wmma_f32_16x16x4_f32(false, a3, false, b0, (short)0, acc30, false, false);
        acc01 = __builtin_amdgcn_wmma_f32_16x16x4_f32(false, a0, false, b1, (short)0, acc01, false, false);
        acc11 = __builtin_amdgcn_wmma_f32_16x16x4_f32(false, a1, false, b1, (short)0, acc11, false, false);
        acc21 = __builtin_amdgcn_wmma_f32_16x16x4_f32(false, a2, false, b1, (short)0, acc21, false, false);
        acc31 = __builtin_amdgcn_wmma_f32_16x16x4_f32(false, a3, false, b1, (short)0, acc31, false, false);
    }

    const float bn0 = bias  ? bias [n0 +  0 + r] : 0.0f;
    const float bn1 = bias  ? bias [n0 + 16 + r] : 0.0f;
    const float sn0 = scale ? scale[n0 +  0 + r] : 1.0f;
    const float sn1 = scale ? scale[n0 + 16 + r] : 1.0f;
    float* Cout0 = C + n0 +  0 + r;
    float* Cout1 = C + n0 + 16 + r;
    #pragma unroll
    for (int v = 0; v < 8; ++v) {
        const int mr = v + 8 * half;
        Cout0[(size_t)(m0 +  0 + mr) * N] = (acc00[v] + bn0) * sn0;
        Cout0[(size_t)(m0 + 16 + mr) * N] = (acc10[v] + bn0) * sn0;
        Cout0[(size_t)(m0 + 32 + mr) * N] = (acc20[v] + bn0) * sn0;
        Cout0[(size_t)(m0 + 48 + mr) * N] = (acc30[v] + bn0) * sn0;
        Cout1[(size_t)(m0 +  0 + mr) * N] = (acc01[v] + bn1) * sn1;
        Cout1[(size_t)(m0 + 16 + mr) * N] = (acc11[v] + bn1) * sn1;
        Cout1[(size_t)(m0 + 32 + mr) * N] = (acc21[v] + bn1) * sn1;
        Cout1[(size_t)(m0 + 48 + mr) * N] = (acc31[v] + bn1) * sn1;
    }
}

// ---------------------------------------------------------------------------
// Depthwise causal conv (K=4) + bias + silu over xBC channels of zxbcdt.
// ---------------------------------------------------------------------------
__global__ void conv_silu_kernel(const float* __restrict__ zxbcdt,
                                 const float* __restrict__ conv_w,
                                 const float* __restrict__ conv_b,
                                 float* __restrict__ xbc)
{
    const int idx = blockIdx.x * blockDim.x + threadIdx.x;
    if (idx >= L_SEQ * CONVDIM) return;
    const int c = idx % CONVDIM;
    const int t = idx / CONVDIM;
    const float* in = zxbcdt + D_SSM + c;
    float s = conv_b[c];
    #pragma unroll
    for (int j = 0; j < KCONV; ++j) {
        const int tt = t - (KCONV - 1) + j;
        if (tt >= 0) s += conv_w[c * KCONV + j] * in[(size_t)tt * PROJDIM];
    }
    xbc[idx] = s / (1.0f + expf(-s));       // silu
}

// ---------------------------------------------------------------------------
// dt = softplus(dt_raw + dt_bias);  a = exp(-dt * exp(A_log))
// ---------------------------------------------------------------------------
__global__ void dt_kernel(const float* __restrict__ zxbcdt,
                          const float* __restrict__ A_log,
                          const float* __restrict__ dt_bias,
                          float* __restrict__ dt_arr,
                          float* __restrict__ a_arr)
{
    const int idx = blockIdx.x * blockDim.x + threadIdx.x;
    if (idx >= L_SEQ * H_HEADS) return;
    const int h = idx % H_HEADS;
    const int t = idx / H_HEADS;
    const float x  = zxbcdt[(size_t)t * PROJDIM + D_SSM + CONVDIM + h] + dt_bias[h];
    const float dt = (x > 20.0f) ? x : log1pf(expf(x));
    dt_arr[idx] = dt;
    a_arr[idx]  = expf(-dt * expf(A_log[h]));
}

// ---------------------------------------------------------------------------
// SSM scan: each (h,p,s) is an independent scalar recurrence.
// grid = H * (P/8) blocks of 256 threads; wave = one p, lane = s.
// y[p] = wave-reduce_s(state * C[s]) + D[h]*x[p]
// ---------------------------------------------------------------------------
__global__ void ssm_scan_kernel(const float* __restrict__ xbc,
                                const float* __restrict__ dt_arr,
                                const float* __restrict__ a_arr,
                                const float* __restrict__ D_param,
                                float* __restrict__ y_raw)
{
    const int lane = threadIdx.x & 31;            // s
    const int wave = threadIdx.x >> 5;            // p within block
    const int h    = blockIdx.x >> 3;
    const int p    = ((blockIdx.x & 7) << 3) + wave;
    const int g    = h >> 2;                      // head -> group (rep = 4)

    const float* xptr = xbc + h * P_DIM + p;
    const float* bptr = xbc + D_SSM + g * S_STATE + lane;
    const float* cptr = xbc + D_SSM + G_GRP * S_STATE + g * S_STATE + lane;
    const float  Dh   = D_param[h];

    float state = 0.0f;
    for (int t = 0; t < L_SEQ; ++t) {
        const float a  = a_arr [t * H_HEADS + h];
        const float dt = dt_arr[t * H_HEADS + h];
        const float xv = xptr[(size_t)t * CONVDIM];
        const float Bv = bptr[(size_t)t * CONVDIM];
        const float Cv = cptr[(size_t)t * CONVDIM];
        state = a * state + dt * xv * Bv;
        float yv = state * Cv;
        #pragma unroll
        for (int off = 16; off > 0; off >>= 1)
            yv += __shfl_xor(yv, off, 32);
        if (lane == 0)
            y_raw[(size_t)t * D_SSM + h * P_DIM + p] = yv + Dh * xv;
    }
}

// ---------------------------------------------------------------------------
// Gate by silu(z) + group RMSNorm (G=8 groups of 256 = one wave32 x 8 elems).
// ---------------------------------------------------------------------------
__global__ void gate_norm_kernel(const float* __restrict__ y_raw,
                                 const float* __restrict__ zxbcdt,
                                 const float* __restrict__ norm_w,
                                 float* __restrict__ y_out)
{
    const int t    = blockIdx.x;
    const int i    = threadIdx.x;       // 0..255
    const int base = i * 8;             // wave w covers elems [256w, 256w+256)
    float yv[8];
    float ss = 0.0f;
    #pragma unroll
    for (int j = 0; j < 8; ++j) {
        const float y = y_raw [(size_t)t * D_SSM   + base + j];
        const float z = zxbcdt[(size_t)t * PROJDIM + base + j];
        const float v = y * (z / (1.0f + expf(-z)));
        yv[j] = v;
        ss += v * v;
    }
    #pragma unroll
    for (int off = 16; off > 0; off >>= 1)
        ss += __shfl_xor(ss, off, 32);
    const float rn = rsqrtf(ss * (1.0f / 256.0f) + 1e-5f);
    #pragma unroll
    for (int j = 0; j < 8; ++j)
        y_out[(size_t)t * D_SSM + base + j] = yv[j] * rn * norm_w[base + j];
}

// ---------------------------------------------------------------------------
extern "C" void kernel_launch(void* const* d_in, const int* in_sizes, int n_in,
                              void* d_out, int out_size, void* d_ws, size_t ws_size,
                              hipStream_t stream)
{
    const float* hs      = (const float*)d_in[0];
    const float* W_in    = (const float*)d_in[1];
    const float* b_in    = (const float*)d_in[2];
    const float* conv_w  = (const float*)d_in[3];
    const float* conv_b  = (const float*)d_in[4];
    const float* A_log   = (const float*)d_in[5];
    const float* D_param = (const float*)d_in[6];
    const float* dt_bias = (const float*)d_in[7];
    const float* norm_w  = (const float*)d_in[8];
    const float* W_out   = (const float*)d_in[9];
    const float* b_out   = (const float*)d_in[10];
    const float* mup     = (const float*)d_in[11];
    float* out = (float*)d_out;

    float* ws = (float*)d_ws;
    float* zxbcdt = ws;  ws += (size_t)L_SEQ * PROJDIM;   // 19.0 MB
    float* xbc    = ws;  ws += (size_t)L_SEQ * CONVDIM;   // 10.5 MB
    float* dt_arr = ws;  ws += (size_t)L_SEQ * H_HEADS;
    float* a_arr  = ws;  ws += (size_t)L_SEQ * H_HEADS;
    float* y_raw  = ws;  ws += (size_t)L_SEQ * D_SSM;     //  8.4 MB
    float* y_nrm  = ws;                                   //  8.4 MB

    // 1) in-proj: zxbcdt = (hs @ W_in + b_in) * mup    [1024 x 4640]
    {
        dim3 grid((PROJDIM + 127) / 128, L_SEQ / 64);
        gemm_wmma_f32<<<grid, 128, 0, stream>>>(hs, W_in, b_in, mup, zxbcdt,
                                                L_SEQ, PROJDIM, DMODEL);
    }
    // 2) depthwise causal conv + silu over xBC          [1024 x 2560]
    conv_silu_kernel<<<(L_SEQ * CONVDIM + 255) / 256, 256, 0, stream>>>(
        zxbcdt, conv_w, conv_b, xbc);
    // 3) dt / a per (t, h)
    dt_kernel<<<(L_SEQ * H_HEADS + 255) / 256, 256, 0, stream>>>(
        zxbcdt, A_log, dt_bias, dt_arr, a_arr);
    // 4) sequential SSM scan (parallel over h, p, s)
    ssm_scan_kernel<<<H_HEADS * (P_DIM / 8), 256, 0, stream>>>(
        xbc, dt_arr, a_arr, D_param, y_raw);
    // 5) gate + group RMSNorm
    gate_norm_kernel<<<L_SEQ, 256, 0, stream>>>(y_raw, zxbcdt, norm_w, y_nrm);
    // 6) out-proj: out = y_nrm @ W_out + b_out          [1024 x 2048]
    {
        dim3 grid(DMODEL / 128, L_SEQ / 64);
        gemm_wmma_f32<<<grid, 128, 0, stream>>>(y_nrm, W_out, b_out, nullptr, out,
                                                L_SEQ, DMODEL, DMODEL);
    }
}